// PolicyLP_PlusSP_66477503808089
// MI455X (gfx1250) — compile-verified
//
#include <hip/hip_runtime.h>
#include <hip/hip_bf16.h>
#include <math.h>

#define NN 64
#define NXE 4096
#define BLK 256
#define NITER 25
#define EPSQ 0.1f
#define SREG 1e-9f
#define BIGV 1e10f

typedef __attribute__((ext_vector_type(2))) float v2f;
typedef __attribute__((ext_vector_type(8))) float v8f;

__device__ __forceinline__ float blk_sum(float v, float* red) {
  int t = threadIdx.x;
  red[t] = v;
  __syncthreads();
  for (int off = BLK / 2; off > 0; off >>= 1) {
    if (t < off) red[t] += red[t + off];
    __syncthreads();
  }
  float r = red[0];
  __syncthreads();
  return r;
}

__device__ __forceinline__ float blk_min(float v, float* red) {
  int t = threadIdx.x;
  red[t] = v;
  __syncthreads();
  for (int off = BLK / 2; off > 0; off >>= 1) {
    if (t < off) red[t] = fminf(red[t], red[t + off]);
    __syncthreads();
  }
  float r = red[0];
  __syncthreads();
  return r;
}

__global__ __launch_bounds__(BLK) void qp_ipm_kernel(
    const float* __restrict__ x, const float* __restrict__ gid,
    const float* __restrict__ expo, float* __restrict__ out) {
  // Persistent per-batch state, all LDS resident (~133 KB < 320 KB/WGP).
  __shared__ float z_s[NXE], s1_s[NXE], s2_s[NXE], l1_s[NXE], l2_s[NXE];
  __shared__ float Dm_s[NXE];    // invD viewed as 64x64 matrix M
  __shared__ float rz_s[NXE];    // rhs_z, recycled as dz
  __shared__ float Tm_s[NN * NN];// Schur-reduced 64x64 / Cholesky factor
  __shared__ float red_s[BLK];
  __shared__ float fv[NN], ev[NN], rr_inv[NN], cc_s[NN], uu[NN], vv[NN];
  __shared__ float br[NN], bc[NN], gg[NN], hh[NN], g0[NN], dyr[NN], dyc[NN];
  __shared__ float yrow[NN], ycol[NN];
  __shared__ float sS1, qS, tS, dyfS, yfairS;

  const int t = threadIdx.x;
  const int b = blockIdx.x;
  const float* p = x + b * NXE;

  // ---- init carry: z=0, y=0, s=1, lam=1 ----
  for (int k = t; k < NXE; k += BLK) {
    z_s[k] = 0.0f; s1_s[k] = 1.0f; s2_s[k] = 1.0f; l1_s[k] = 1.0f; l2_s[k] = 1.0f;
  }
  if (t < NN) { yrow[t] = 0.0f; ycol[t] = 0.0f; ev[t] = expo[t]; }
  if (t == 0) {
    yfairS = 0.0f;
    float sg = 0.0f;
    for (int i = 0; i < NN; ++i) sg += gid[i];  // group_ids[0] row only (matches ref)
    sS1 = sg;
  }
  __syncthreads();
  if (t < NN) {
    float g = gid[t];
    fv[t] = g / sS1 - (1.0f - g) / (64.0f - sS1);
  }
  __syncthreads();

  for (int iter = 0; iter < NITER; ++iter) {
    // ---- mu = sum(s*lam)/m ----
    float part = 0.0f;
    for (int k = t; k < NXE; k += BLK)
      part += s1_s[k] * l1_s[k] + s2_s[k] * l2_s[k];
    float mu = blk_sum(part, red_s) * (1.0f / 8192.0f);
    float smu = 0.1f * mu;
    float yf = yfairS;

    // ---- rhs_z, invD, fair component of r_eq ----
    float pzf = 0.0f;
    for (int k = t; k < NXE; k += BLK) {
      int i = k >> 6, j = k & 63;
      float fe = fv[i] * ev[j];
      float zz = z_s[k], ss1 = s1_s[k], ss2 = s2_s[k], ll1 = l1_s[k], ll2 = l2_s[k];
      float rdual = EPSQ * zz + p[k] + yrow[i] + ycol[j] + yf * fe + (ll2 - ll1);
      float r1 = ss1 - zz;            // r_ineq lower
      float r2 = zz + ss2 - 1.0f;     // r_ineq upper
      float tmp1 = (ll1 * r1 - (ll1 * ss1 - smu)) / ss1;
      float tmp2 = (ll2 * r2 - (ll2 * ss2 - smu)) / ss2;
      rz_s[k] = -(rdual - tmp1 + tmp2);
      Dm_s[k] = 1.0f / (EPSQ + ll1 / ss1 + ll2 / ss2);
      pzf += fe * zz;
    }
    __syncthreads();
    float reqf = blk_sum(pzf, red_s);

    // ---- fair rhs_y component and S[fair,fair] ----
    float pwf = 0.0f, psc = 0.0f;
    for (int k = t; k < NXE; k += BLK) {
      int i = k >> 6, j = k & 63;
      float fe = fv[i] * ev[j];
      float d = Dm_s[k];
      pwf += rz_s[k] * d * fe;
      psc += fe * fe * d;
    }
    float bfair = blk_sum(pwf, red_s) + reqf;
    float sc = blk_sum(psc, red_s) + SREG;

    // ---- row (t<64) and col (64<=t<128) scalars of arrow matrix + rhs ----
    if (t < NN) {
      int i = t;
      float rs = 0.0f, ue = 0.0f, zrs = 0.0f, wrs = 0.0f;
      for (int j = 0; j < NN; ++j) {
        float d = Dm_s[i * NN + j];
        rs += d;
        ue += d * ev[j];
        zrs += z_s[i * NN + j];
        wrs += rz_s[i * NN + j] * d;
      }
      rr_inv[i] = 1.0f / (rs + SREG);
      uu[i] = fv[i] * ue;
      br[i] = wrs + zrs - 1.0f;       // rowsum(w) + r_eq_row
    } else if (t < 2 * NN) {
      int j = t - NN;
      float cs = 0.0f, fd = 0.0f, zcs = 0.0f, wcs = 0.0f;
      for (int i = 0; i < NN; ++i) {
        float d = Dm_s[i * NN + j];
        cs += d;
        fd += fv[i] * d;
        zcs += z_s[i * NN + j];
        wcs += rz_s[i * NN + j] * d;
      }
      cc_s[j] = cs + SREG;
      vv[j] = ev[j] * fd;
      bc[j] = wcs + zcs - 1.0f;       // colsum(w) + r_eq_col
    }
    __syncthreads();

    // ---- WMMA: P = (Dr^-1 M)^T * M  (64x64x64 f32), Tm = diag(c) - P ----
    {
      int wave = t >> 5, lane = t & 31;
      int m16 = lane & 15;
      int khalf = (lane >> 4) * 2;     // ISA 16x4 f32 A layout: VGPRs hold K pair
      for (int tt = wave * 2; tt < wave * 2 + 2; ++tt) {  // 16 tiles, 2 per wave
        int tm = tt >> 2, tn = tt & 3;
        v8f acc = {0.f, 0.f, 0.f, 0.f, 0.f, 0.f, 0.f, 0.f};
        for (int kb = 0; kb < NN; kb += 4) {
          int k0 = kb + khalf, k1 = k0 + 1;
          v2f afr, bfr;
          afr.x = Dm_s[k0 * NN + tm * 16 + m16] * rr_inv[k0]; // A[m][k]=M[k][m]/r[k]
          afr.y = Dm_s[k1 * NN + tm * 16 + m16] * rr_inv[k1];
          bfr.x = Dm_s[k0 * NN + tn * 16 + m16];              // B[k][n]=M[k][n]
          bfr.y = Dm_s[k1 * NN + tn * 16 + m16];
          acc = __builtin_amdgcn_wmma_f32_16x16x4_f32(
              false, afr, false, bfr, (short)0, acc, false, false);
        }
        int col = tn * 16 + m16;
        int rbase = tm * 16 + 8 * (lane >> 4);  // C/D: VGPR v -> row v + 8*(L/16)
        #pragma unroll
        for (int v = 0; v < 8; ++v) {
          int row = rbase + v;
          float dg = (row == col) ? cc_s[col] : 0.0f;
          Tm_s[row * NN + col] = dg - acc[v];
        }
      }
    }
    // ---- g = v - M^T(u/r), h = b_c - M^T(b_r/r) ----
    if (t < NN) {
      int j = t;
      float a1 = 0.0f, a2 = 0.0f;
      for (int i = 0; i < NN; ++i) {
        float d = Dm_s[i * NN + j] * rr_inv[i];
        a1 += d * uu[i];
        a2 += d * br[i];
      }
      float gv = vv[j] - a1;
      gg[j] = gv; g0[j] = gv;
      hh[j] = bc[j] - a2;
    }
    __syncthreads();
    if (t == 0) {  // bordered scalar row
      float q = sc, ts = bfair;
      for (int i = 0; i < NN; ++i) {
        q  -= uu[i] * uu[i] * rr_inv[i];
        ts -= uu[i] * br[i] * rr_inv[i];
      }
      qS = q; tS = ts;
    }
    __syncthreads();

    // ---- Cholesky of Tm (lower, in place, pivot-clamped) ----
    for (int kk = 0; kk < NN; ++kk) {
      if (t == 0) Tm_s[kk * NN + kk] = sqrtf(fmaxf(Tm_s[kk * NN + kk], 1e-20f));
      __syncthreads();
      if (t > kk && t < NN) Tm_s[t * NN + kk] /= Tm_s[kk * NN + kk];
      __syncthreads();
      if (t > kk && t < NN) {
        float lik = Tm_s[t * NN + kk];
        for (int j = kk + 1; j <= t; ++j)
          Tm_s[t * NN + j] -= lik * Tm_s[j * NN + kk];
      }
      __syncthreads();
    }

    // ---- two triangular solves: w1 <- T^-1 g (in gg), w2 <- T^-1 h (in hh) ----
    if (t < 2) {
      float* xv = (t == 0) ? gg : hh;
      for (int kk = 0; kk < NN; ++kk) {
        float a = xv[kk];
        for (int j = 0; j < kk; ++j) a -= Tm_s[kk * NN + j] * xv[j];
        xv[kk] = a / Tm_s[kk * NN + kk];
      }
      for (int kk = NN - 1; kk >= 0; --kk) {
        float a = xv[kk];
        for (int j = kk + 1; j < NN; ++j) a -= Tm_s[j * NN + kk] * xv[j];
        xv[kk] = a / Tm_s[kk * NN + kk];
      }
    }
    __syncthreads();
    if (t == 0) {
      float gw1 = 0.0f, gw2 = 0.0f;
      for (int j = 0; j < NN; ++j) { gw1 += g0[j] * gg[j]; gw2 += g0[j] * hh[j]; }
      dyfS = (tS - gw2) / (qS - gw1);
    }
    __syncthreads();
    float dyf = dyfS;
    if (t < NN) dyc[t] = hh[t] - dyf * gg[t];
    __syncthreads();
    if (t < NN) {
      float a = br[t] - uu[t] * dyf;
      for (int j = 0; j < NN; ++j) a -= Dm_s[t * NN + j] * dyc[j];
      dyr[t] = a * rr_inv[t];
    }
    __syncthreads();

    // ---- dz, step length ----
    float amin = BIGV;
    for (int k = t; k < NXE; k += BLK) {
      int i = k >> 6, j = k & 63;
      float fe = fv[i] * ev[j];
      float dz = (rz_s[k] - (dyr[i] + dyc[j] + dyf * fe)) * Dm_s[k];
      float zz = z_s[k], ss1 = s1_s[k], ss2 = s2_s[k], ll1 = l1_s[k], ll2 = l2_s[k];
      float ds1 = zz - ss1 + dz;
      float ds2 = 1.0f - zz - ss2 - dz;
      float dl1 = ((smu - ll1 * ss1) - ll1 * ds1) / ss1;
      float dl2 = ((smu - ll2 * ss2) - ll2 * ds2) / ss2;
      if (ds1 < 0.0f) amin = fminf(amin, -ss1 / ds1);
      if (ds2 < 0.0f) amin = fminf(amin, -ss2 / ds2);
      if (dl1 < 0.0f) amin = fminf(amin, -ll1 / dl1);
      if (dl2 < 0.0f) amin = fminf(amin, -ll2 / dl2);
      rz_s[k] = dz;  // recycle rhs_z storage as dz
    }
    float alpha = fminf(1.0f, 0.99f * blk_min(amin, red_s));

    // ---- apply update ----
    for (int k = t; k < NXE; k += BLK) {
      float dz = rz_s[k];
      float zz = z_s[k], ss1 = s1_s[k], ss2 = s2_s[k], ll1 = l1_s[k], ll2 = l2_s[k];
      float ds1 = zz - ss1 + dz;
      float ds2 = 1.0f - zz - ss2 - dz;
      float dl1 = ((smu - ll1 * ss1) - ll1 * ds1) / ss1;
      float dl2 = ((smu - ll2 * ss2) - ll2 * ds2) / ss2;
      z_s[k]  = zz + alpha * dz;
      s1_s[k] = ss1 + alpha * ds1;
      s2_s[k] = ss2 + alpha * ds2;
      l1_s[k] = ll1 + alpha * dl1;
      l2_s[k] = ll2 + alpha * dl2;
    }
    if (t < NN) { yrow[t] += alpha * dyr[t]; ycol[t] += alpha * dyc[t]; }
    if (t == 0) yfairS += alpha * dyf;
    __syncthreads();
  }

  for (int k = t; k < NXE; k += BLK)
    out[b * NXE + k] = z_s[k];
}

extern "C" void kernel_launch(void* const* d_in, const int* in_sizes, int n_in,
                              void* d_out, int out_size, void* d_ws, size_t ws_size,
                              hipStream_t stream) {
  (void)in_sizes; (void)n_in; (void)d_ws; (void)ws_size; (void)out_size;
  const float* x    = (const float*)d_in[0];   // (32, 4096) f32
  const float* gid  = (const float*)d_in[1];   // (32, 64)   f32
  const float* expo = (const float*)d_in[2];   // (64,)      f32
  qp_ipm_kernel<<<32, BLK, 0, stream>>>(x, gid, expo, (float*)d_out);
}